// GNNNet_89034672046438
// MI455X (gfx1250) — compile-verified
//
#include <hip/hip_runtime.h>
#include <hip/hip_bf16.h>

// ---------------- problem constants (from reference) ----------------
#define NN      60000            // nodes (== 16 * 3750, no partial node tiles)
#define NE      300000           // edges
#define FIN     13
#define FOUT    64
#define KSPL    5                // kernel size per dim
#define NW      125              // K^3 weight matrices
#define NG      256              // graphs
#define NPAIRS  (NE * 8)         // (edge, corner) pairs
#define AGG_ROWS (NN + 16)       // +16 dump rows for branch-free atomic scatter

typedef __attribute__((ext_vector_type(2))) float v2f;
typedef __attribute__((ext_vector_type(8))) float v8f;

// ---------------------------------------------------------------
// K0: zero scratch region (agg+dump, deg, pooledSum, cnt, bins)
// ---------------------------------------------------------------
__global__ void k_zero(float* p, long n) {
    long i = (long)blockIdx.x * blockDim.x + threadIdx.x;
    long stride = (long)gridDim.x * blockDim.x;
    for (; i < n; i += stride) p[i] = 0.0f;
}

// helper: per-edge spline indices
__device__ __forceinline__ void spline_corners(const float* __restrict__ pseudo,
                                               int e, int* i0, float* fr) {
    #pragma unroll
    for (int d = 0; d < 3; ++d) {
        float v = pseudo[e * 3 + d] * (float)(KSPL - 1);
        float f = floorf(v);
        f = fminf(fmaxf(f, 0.0f), (float)(KSPL - 2));
        i0[d] = (int)f;
        fr[d] = v - f;
    }
}

// ---------------------------------------------------------------
// K1: histogram pairs by weight index + target degree
// ---------------------------------------------------------------
__global__ void k_edge_hist(const float* __restrict__ pseudo,
                            const int* __restrict__ ei,
                            unsigned* __restrict__ binCount,
                            float* __restrict__ deg) {
    int e = blockIdx.x * blockDim.x + threadIdx.x;
    if (e >= NE) return;
    atomicAdd(&deg[ei[NE + e]], 1.0f);
    int i0[3]; float fr[3];
    spline_corners(pseudo, e, i0, fr);
    #pragma unroll
    for (int s = 0; s < 8; ++s) {
        int s0 = s & 1, s1 = (s >> 1) & 1, s2 = (s >> 2) & 1;
        int wi = (i0[0] + s0) + (i0[1] + s1) * KSPL + (i0[2] + s2) * KSPL * KSPL;
        atomicAdd(&binCount[wi], 1u);
    }
}

// ---------------------------------------------------------------
// K2: exclusive scan over 125 bins (tiny; serial on lane 0)
// ---------------------------------------------------------------
__global__ void k_bin_scan(const unsigned* __restrict__ binCount,
                           unsigned* __restrict__ binStart,
                           unsigned* __restrict__ binCursor) {
    if (threadIdx.x == 0) {
        unsigned run = 0;
        for (int i = 0; i < NW; ++i) {
            binStart[i]  = run;
            binCursor[i] = run;
            run += binCount[i];
        }
    }
}

// ---------------------------------------------------------------
// K3: scatter (edge, corner) records into bins
// ---------------------------------------------------------------
__global__ void k_edge_scatter(const float* __restrict__ pseudo,
                               const int* __restrict__ ei,
                               unsigned* __restrict__ binCursor,
                               unsigned* __restrict__ pairEdge,
                               float* __restrict__ pairB) {
    int e = blockIdx.x * blockDim.x + threadIdx.x;
    if (e >= NE) return;
    int i0[3]; float fr[3];
    spline_corners(pseudo, e, i0, fr);
    #pragma unroll
    for (int s = 0; s < 8; ++s) {
        int s0 = s & 1, s1 = (s >> 1) & 1, s2 = (s >> 2) & 1;
        float b = (s0 ? fr[0] : 1.0f - fr[0]) *
                  (s1 ? fr[1] : 1.0f - fr[1]) *
                  (s2 ? fr[2] : 1.0f - fr[2]);
        int wi = (i0[0] + s0) + (i0[1] + s1) * KSPL + (i0[2] + s2) * KSPL * KSPL;
        unsigned idx = atomicAdd(&binCursor[wi], 1u);
        pairEdge[idx] = (unsigned)e;
        pairB[idx] = b;
    }
}

// ---------------------------------------------------------------
// K4: per-bin dense GEMM via V_WMMA_F32_16X16X4_F32
//     A[16 pairs x 16(K pad)] = b * x[src]  (LDS, stride 17)
//     B = W[bin] [16(K pad) x 64] in LDS
//     branch-free global_atomic_add_f32 scatter into agg[tgt]
//     (invalid rows target dump row NN; never read back)
// ---------------------------------------------------------------
__global__ __launch_bounds__(128) void k_spline_gemm(
        const float* __restrict__ x,
        const int* __restrict__ ei,
        const float* __restrict__ W,
        const unsigned* __restrict__ binCount,
        const unsigned* __restrict__ binStart,
        const unsigned* __restrict__ pairEdge,
        const float* __restrict__ pairB,
        float* __restrict__ agg) {
    __shared__ float Wl[16 * 64];        // B tile, K padded to 16
    __shared__ float Al[4][16 * 17];     // per-wave A tile, stride-17 pad
    __shared__ int   Tl[4][16];          // per-wave target node per row

    const int w = blockIdx.x;            // bin id
    for (int i = threadIdx.x; i < 16 * 64; i += blockDim.x) {
        int k = i >> 6, n = i & 63;
        Wl[i] = (k < FIN) ? W[(w * FIN + k) * FOUT + n] : 0.0f;
    }
    __syncthreads();

    const unsigned start = binStart[w];
    const unsigned count = binCount[w];
    const unsigned ntile = (count + 15u) >> 4;

    const int wave = threadIdx.x >> 5;
    const int lane = threadIdx.x & 31;
    const unsigned waveG = blockIdx.y * (blockDim.x >> 5) + wave;
    const unsigned waveS = gridDim.y * (blockDim.x >> 5);

    float* A = &Al[wave][0];
    int*   T = &Tl[wave][0];
    const int row = lane & 15;
    const int hi  = (lane >= 16);

    for (unsigned t = waveG; t < ntile; t += waveS) {
        // ---- stage A tile + row metadata ----
        unsigned pidx = start + t * 16u + (unsigned)row;
        int   e  = -1; float bco = 0.0f; int tg = NN;     // dump row if invalid
        if (pidx < start + count) {
            e   = (int)pairEdge[pidx];
            bco = pairB[pidx];
            tg  = ei[NE + e];
        }
        int srcn = (e >= 0) ? ei[e] : 0;
        int cbase = hi ? 8 : 0;
        #pragma unroll
        for (int c = 0; c < 8; ++c) {
            int col = cbase + c;
            float v = 0.0f;
            if (e >= 0 && col < FIN) v = bco * x[srcn * FIN + col];
            A[row * 17 + col] = v;
        }
        if (!hi) T[row] = tg;
        asm volatile("s_wait_dscnt 0" ::: "memory");  // wave-local LDS fence

        // hoist target addresses: lane handles rows M = hi*8 + (0..7)
        const int4* Tv = (const int4*)&T[hi * 8];
        int4 t0 = Tv[0], t1 = Tv[1];
        int abase[8];
        abase[0] = t0.x * FOUT + row;  abase[1] = t0.y * FOUT + row;
        abase[2] = t0.z * FOUT + row;  abase[3] = t0.w * FOUT + row;
        abase[4] = t1.x * FOUT + row;  abase[5] = t1.y * FOUT + row;
        abase[6] = t1.z * FOUT + row;  abase[7] = t1.w * FOUT + row;

        // ---- 4 N-tiles x 4 K-steps of fp32 WMMA + dense atomic scatter ----
        for (int nt = 0; nt < 4; ++nt) {
            v8f acc = {};
            #pragma unroll
            for (int kk = 0; kk < 4; ++kk) {
                int kbase = kk * 4 + (hi ? 2 : 0);  // lanes 0-15: K=0,1 ; 16-31: K=2,3
                v2f af, bf;
                af.x = A[row * 17 + kbase];
                af.y = A[row * 17 + kbase + 1];
                int n = nt * 16 + row;
                bf.x = Wl[kbase * 64 + n];
                bf.y = Wl[(kbase + 1) * 64 + n];
                acc = __builtin_amdgcn_wmma_f32_16x16x4_f32(
                        false, af, false, bf, (short)0, acc, false, false);
            }
            #pragma unroll
            for (int r = 0; r < 8; ++r)
                atomicAdd(&agg[abase[r] + nt * 16], acc[r]);
        }
    }
}

// ---------------------------------------------------------------
// K5: node update h = agg/deg + x@root (WMMA) + bias, ELU,
//     atomic pooling per graph. NN % 16 == 0 -> no edge cases.
// ---------------------------------------------------------------
__global__ __launch_bounds__(128) void k_node_update(
        const float* __restrict__ x,
        const float* __restrict__ root,
        const float* __restrict__ bias,
        const int* __restrict__ batchv,
        const float* __restrict__ agg,
        const float* __restrict__ deg,
        float* __restrict__ pooledSum,
        float* __restrict__ cnt) {
    __shared__ float Rl[16 * 64];
    __shared__ float Bl[64];
    __shared__ float Al[4][16 * 17];
    __shared__ int   Gl[4][16];          // per-wave graph id per row

    for (int i = threadIdx.x; i < 16 * 64; i += blockDim.x) {
        int k = i >> 6, n = i & 63;
        Rl[i] = (k < FIN) ? root[k * FOUT + n] : 0.0f;
    }
    if (threadIdx.x < FOUT) Bl[threadIdx.x] = bias[threadIdx.x];
    __syncthreads();

    const int wave = threadIdx.x >> 5;
    const int lane = threadIdx.x & 31;
    const int hi   = (lane >= 16);
    const int row  = lane & 15;
    const int t    = blockIdx.x * (blockDim.x >> 5) + wave;   // node tile
    const int base = t * 16;
    if (base >= NN) return;                                    // wave-uniform exit

    float* A = &Al[wave][0];
    int*   Gm = &Gl[wave][0];
    int node0 = base + row;
    int cbase = hi ? 8 : 0;
    #pragma unroll
    for (int c = 0; c < 8; ++c) {
        int col = cbase + c;
        A[row * 17 + col] = (col < FIN) ? x[node0 * FIN + col] : 0.0f;
    }
    if (!hi) {
        int g = batchv[node0];
        Gm[row] = g;
        atomicAdd(&cnt[g], 1.0f);        // one count per node
    }
    asm volatile("s_wait_dscnt 0" ::: "memory");

    // hoist per-row graph ids and inverse degree; rows M = hi*8 + (0..7)
    const int4* Gv = (const int4*)&Gm[hi * 8];
    int4 g0 = Gv[0], g1 = Gv[1];
    int pbase[8];
    pbase[0] = g0.x * FOUT + row;  pbase[1] = g0.y * FOUT + row;
    pbase[2] = g0.z * FOUT + row;  pbase[3] = g0.w * FOUT + row;
    pbase[4] = g1.x * FOUT + row;  pbase[5] = g1.y * FOUT + row;
    pbase[6] = g1.z * FOUT + row;  pbase[7] = g1.w * FOUT + row;
    float idg[8];
    #pragma unroll
    for (int r = 0; r < 8; ++r)
        idg[r] = 1.0f / fmaxf(deg[base + hi * 8 + r], 1.0f);

    for (int nt = 0; nt < 4; ++nt) {
        v8f acc = {};
        #pragma unroll
        for (int kk = 0; kk < 4; ++kk) {
            int kbase = kk * 4 + (hi ? 2 : 0);
            v2f af, bf;
            af.x = A[row * 17 + kbase];
            af.y = A[row * 17 + kbase + 1];
            int n = nt * 16 + row;
            bf.x = Rl[kbase * 64 + n];
            bf.y = Rl[(kbase + 1) * 64 + n];
            acc = __builtin_amdgcn_wmma_f32_16x16x4_f32(
                    false, af, false, bf, (short)0, acc, false, false);
        }
        int col = nt * 16 + row;
        float bcol = Bl[col];
        #pragma unroll
        for (int r = 0; r < 8; ++r) {
            int node = base + hi * 8 + r;
            float h = acc[r] + agg[node * FOUT + col] * idg[r] + bcol;
            h = (h > 0.0f) ? h : expm1f(h);                // ELU(alpha=1)
            atomicAdd(&pooledSum[pbase[r] + nt * 16], h);
        }
    }
}

// ---------------------------------------------------------------
// K6: mean-pool finish, BatchNorm (batch stats), head, L2-normalize
// ---------------------------------------------------------------
__global__ void k_head(const float* __restrict__ pooledSum,
                       const float* __restrict__ cnt,
                       const float* __restrict__ gamma,
                       const float* __restrict__ beta,
                       const float* __restrict__ lin_w,
                       const float* __restrict__ lin_b,
                       float* __restrict__ out) {
    __shared__ float mu[FOUT], inv[FOUT];
    int tid = threadIdx.x;
    if (tid < FOUT) {
        float s = 0.0f, s2 = 0.0f;
        for (int g = 0; g < NG; ++g) {
            float p = pooledSum[g * FOUT + tid] / fmaxf(cnt[g], 1.0f);
            s += p; s2 += p * p;
        }
        float m = s / (float)NG;
        float v = s2 / (float)NG - m * m;
        mu[tid]  = m;
        inv[tid] = rsqrtf(v + 1e-5f);
    }
    __syncthreads();
    int g = tid;                       // 256 threads == NG graphs
    float o0 = lin_b[0], o1 = lin_b[1], o2 = lin_b[2];
    float ic = 1.0f / fmaxf(cnt[g], 1.0f);
    for (int f = 0; f < FOUT; ++f) {
        float p  = pooledSum[g * FOUT + f] * ic;
        float pb = (p - mu[f]) * inv[f] * gamma[f] + beta[f];
        o0 += pb * lin_w[f * 3 + 0];
        o1 += pb * lin_w[f * 3 + 1];
        o2 += pb * lin_w[f * 3 + 2];
    }
    float nrm = sqrtf(o0 * o0 + o1 * o1 + o2 * o2);
    float s = 1.0f / fmaxf(nrm, 1e-12f);
    out[g * 3 + 0] = o0 * s;
    out[g * 3 + 1] = o1 * s;
    out[g * 3 + 2] = o2 * s;
}

// ---------------------------------------------------------------
extern "C" void kernel_launch(void* const* d_in, const int* in_sizes, int n_in,
                              void* d_out, int out_size, void* d_ws, size_t ws_size,
                              hipStream_t stream) {
    const int*   batchv = (const int*)  d_in[0];
    const float* x      = (const float*)d_in[1];
    const int*   ei     = (const int*)  d_in[2];
    const float* pseudo = (const float*)d_in[3];
    const float* W      = (const float*)d_in[4];
    const float* root   = (const float*)d_in[5];
    const float* bias   = (const float*)d_in[6];
    const float* gamma  = (const float*)d_in[7];
    const float* beta   = (const float*)d_in[8];
    const float* lin_w  = (const float*)d_in[9];
    const float* lin_b  = (const float*)d_in[10];
    float* out = (float*)d_out;

    // workspace layout (~35 MB)
    float*    ws        = (float*)d_ws;
    float*    agg       = ws;                              // AGG_ROWS*FOUT
    float*    deg       = agg + (long)AGG_ROWS * FOUT;     // NN
    float*    pooledSum = deg + NN;                        // NG*FOUT
    float*    cnt       = pooledSum + NG * FOUT;           // NG
    unsigned* binCount  = (unsigned*)(cnt + NG);           // 128
    unsigned* binStart  = binCount + 128;                  // 128
    unsigned* binCursor = binStart + 128;                  // 128
    unsigned* pairEdge  = binCursor + 128;                 // NPAIRS
    float*    pairB     = (float*)(pairEdge + NPAIRS);     // NPAIRS

    long zeroN = (long)AGG_ROWS * FOUT + NN + NG * FOUT + NG + 3 * 128;
    k_zero<<<2048, 256, 0, stream>>>(ws, zeroN);

    k_edge_hist<<<(NE + 255) / 256, 256, 0, stream>>>(pseudo, ei, binCount, deg);
    k_bin_scan<<<1, 32, 0, stream>>>(binCount, binStart, binCursor);
    k_edge_scatter<<<(NE + 255) / 256, 256, 0, stream>>>(pseudo, ei, binCursor,
                                                         pairEdge, pairB);
    // 125 bins x 24 tile-groups x 4 waves of fp32-WMMA GEMM
    k_spline_gemm<<<dim3(NW, 24), 128, 0, stream>>>(x, ei, W, binCount, binStart,
                                                    pairEdge, pairB, agg);
    int ntiles = NN / 16;                                  // 3750, exact
    k_node_update<<<(ntiles + 3) / 4, 128, 0, stream>>>(x, root, bias, batchv,
                                                        agg, deg, pooledSum, cnt);
    k_head<<<1, 256, 0, stream>>>(pooledSum, cnt, gamma, beta, lin_w, lin_b, out);
}